// MatrixAttention_18726057410620
// MI455X (gfx1250) — compile-verified
//
#include <hip/hip_runtime.h>
#include <cstdint>

typedef __attribute__((ext_vector_type(2))) float v2f;
typedef __attribute__((ext_vector_type(8))) float v8f;

#define R1 2048
#define R2 2048
#define DDIM 256
#define KBLK 32
#define LDSP 36  // 32 + 4 pad: 16-lane stride-36-dword reads hit 16 distinct banks
#define TILE_ELEMS (128 * LDSP)

#if defined(__has_builtin)
#if __has_builtin(__builtin_amdgcn_global_load_async_to_lds_b128)
#define USE_ASYNC 1
#else
#define USE_ASYNC 0
#endif
#else
#define USE_ASYNC 0
#endif

// Builtin prototype (from clang diagnostic): pointee type is vector_size(16) int,
// with addrspace(1) source and addrspace(3) LDS destination.
typedef int v4i_vs __attribute__((vector_size(16)));
typedef __attribute__((address_space(1))) v4i_vs gv4i;
typedef __attribute__((address_space(3))) v4i_vs lv4i;

// Build AS pointers via integer casts (no addrspacecast needed):
// generic LDS address low 32 bits == LDS byte offset (ISA aperture rules).
__device__ __forceinline__ gv4i* to_global(const void* p) {
    return (gv4i*)(uintptr_t)p;
}
__device__ __forceinline__ lv4i* to_lds(const void* p) {
    return (lv4i*)(uint32_t)(uintptr_t)p;
}

__device__ __forceinline__ void wait_async0() {
#if defined(__has_builtin) && __has_builtin(__builtin_amdgcn_s_wait_asynccnt)
    __builtin_amdgcn_s_wait_asynccnt(0);
#else
    asm volatile("s_wait_asynccnt 0" ::: "memory");
#endif
}

// WG: 128 threads = 4 waves. WG tile 128x128, wave tile 64x64 (4x4 WMMA tiles).
// Ping-pong LDS buffers; async prefetch of block kb+1 overlaps WMMA on block kb.
__global__ __launch_bounds__(128)
void batched_gemm_nt_f32_wmma(const float* __restrict__ m1,
                              const float* __restrict__ m2,
                              float* __restrict__ out) {
    __shared__ float As[2][TILE_ELEMS];
    __shared__ float Bs[2][TILE_ELEMS];

    const int b  = blockIdx.z;
    const int i0 = blockIdx.y * 128;   // row tile in R1
    const int j0 = blockIdx.x * 128;   // col tile in R2

    const int lane = threadIdx.x & 31;
    const int lr   = lane & 15;        // M/N index within 16-wide tile
    const int lh   = lane >> 4;        // selects K-pair (frags) / row-half (C)
    const int w    = threadIdx.x >> 5; // wave id 0..3
    const int wm   = (w >> 1) * 64;    // wave row offset in WG tile
    const int wn   = (w & 1) * 64;     // wave col offset in WG tile

    const float* a_base = m1 + (size_t)b * R1 * DDIM + (size_t)i0 * DDIM;
    const float* b_base = m2 + (size_t)b * R2 * DDIM + (size_t)j0 * DDIM;

    v8f acc[4][4];
#pragma unroll
    for (int mi = 0; mi < 4; ++mi)
#pragma unroll
        for (int ni = 0; ni < 4; ++ni)
            acc[mi][ni] = (v8f){0.f, 0.f, 0.f, 0.f, 0.f, 0.f, 0.f, 0.f};

    // Stage a 128xKBLK slice of A and B into LDS buffer `buf`.
    // 1024 float4s; 8 per thread; 8 consecutive threads cover one 128B row chunk.
    auto stage = [&](int kb, int buf) {
#pragma unroll
        for (int it = 0; it < 8; ++it) {
            const int ff  = (int)threadIdx.x + 128 * it;  // 0..1023
            const int row = ff >> 3;                      // 0..127
            const int c4  = (ff & 7) << 2;                // 0,4,...,28
            const float* ga = a_base + (size_t)row * DDIM + kb + c4;
            const float* gb = b_base + (size_t)row * DDIM + kb + c4;
            float* la = &As[buf][row * LDSP + c4];        // 144B row stride: 16B aligned
            float* lb = &Bs[buf][row * LDSP + c4];
#if USE_ASYNC
            __builtin_amdgcn_global_load_async_to_lds_b128(to_global(ga), to_lds(la), 0, 0);
            __builtin_amdgcn_global_load_async_to_lds_b128(to_global(gb), to_lds(lb), 0, 0);
#else
            *(float4*)la = *(const float4*)ga;
            *(float4*)lb = *(const float4*)gb;
#endif
        }
    };

    // 8 K-steps of 4, 16 WMMA tiles each -> 128 v_wmma per staged block.
    auto compute = [&](int buf) {
        const float* Ap = &As[buf][0];
        const float* Bp = &Bs[buf][0];
#pragma unroll
        for (int kk = 0; kk < KBLK; kk += 4) {
            v2f afr[4], bfr[4];
#pragma unroll
            for (int t = 0; t < 4; ++t) {
                // A 16x4 frag: lane holds row (lr), K = kk + 2*lh + {0,1}
                afr[t] = *(const v2f*)&Ap[(wm + 16 * t + lr) * LDSP + kk + 2 * lh];
                // B 4x16 frag: lane holds col (lr), K = kk + 2*lh + {0,1}
                bfr[t] = *(const v2f*)&Bp[(wn + 16 * t + lr) * LDSP + kk + 2 * lh];
            }
#pragma unroll
            for (int mi = 0; mi < 4; ++mi)
#pragma unroll
                for (int ni = 0; ni < 4; ++ni)
                    acc[mi][ni] = __builtin_amdgcn_wmma_f32_16x16x4_f32(
                        false, afr[mi], false, bfr[ni],
                        (short)0, acc[mi][ni], false, false);
        }
    };

    // Prologue: fill buffer 0.
    stage(0, 0);
#if USE_ASYNC
    wait_async0();
#endif
    __syncthreads();

    int p = 0;
    for (int kb = 0; kb < DDIM; kb += KBLK) {
        if (kb + KBLK < DDIM) stage(kb + KBLK, p ^ 1);  // prefetch next block
        compute(p);
#if USE_ASYNC
        wait_async0();   // my async writes to buf p^1 have landed in LDS
#endif
        __syncthreads(); // all waves: done reading p, done writing p^1
        p ^= 1;
    }

    // Store: C/D layout: VGPR r -> row r + 8*lh, col lr within each 16x16 tile.
    // Non-temporal: streamed 268MB output shouldn't evict reused A/B from L2.
    float* outb = out + (size_t)b * R1 * R2;
#pragma unroll
    for (int mi = 0; mi < 4; ++mi) {
#pragma unroll
        for (int ni = 0; ni < 4; ++ni) {
            const int row0 = i0 + wm + 16 * mi + 8 * lh;
            const int col  = j0 + wn + 16 * ni + lr;
#pragma unroll
            for (int r = 0; r < 8; ++r)
                __builtin_nontemporal_store(acc[mi][ni][r],
                                            &outb[(size_t)(row0 + r) * R2 + col]);
        }
    }
}

extern "C" void kernel_launch(void* const* d_in, const int* in_sizes, int n_in,
                              void* d_out, int out_size, void* d_ws, size_t ws_size,
                              hipStream_t stream) {
    const float* m1 = (const float*)d_in[0];  // [16, 2048, 256] f32
    const float* m2 = (const float*)d_in[1];  // [16, 2048, 256] f32
    float* out = (float*)d_out;               // [16, 2048, 2048] f32

    dim3 grid(R2 / 128, R1 / 128, 16);
    dim3 block(128, 1, 1);
    batched_gemm_nt_f32_wmma<<<grid, block, 0, stream>>>(m1, m2, out);
}